// GATConv_no_fusion_67508295958894
// MI455X (gfx1250) — compile-verified
//
#include <hip/hip_runtime.h>
#include <hip/hip_bf16.h>
#include <math.h>

typedef float v2f __attribute__((ext_vector_type(2)));
typedef float v8f __attribute__((ext_vector_type(8)));
typedef int   v4i __attribute__((vector_size(16)));   // matches builtin's V4i param

#define IN_F   256          // input feats
#define OUT_F  256          // H*D
#define NHEAD  4
#define HDIM   64
#define NEG_SLOPE 0.2f

#define AS1 __attribute__((address_space(1)))
#define AS3 __attribute__((address_space(3)))

#if defined(__has_builtin)
#if __has_builtin(__builtin_amdgcn_global_load_async_to_lds_b128) && \
    __has_builtin(__builtin_amdgcn_s_wait_asynccnt)
#define HAVE_ASYNC_LDS 1
#endif
#endif

#ifdef HAVE_ASYNC_LDS
// global -> LDS async 128-bit transfer, per lane (ASYNCcnt-tracked)
#define GASYNC_B128(gp, lp) \
  __builtin_amdgcn_global_load_async_to_lds_b128((AS1 v4i*)(gp), (AS3 v4i*)(lp), 0, 0)
#endif

// ---------------- Kernel 0: transpose W (256x256) -> Wt so GEMM B loads are clean
__global__ __launch_bounds__(256) void k_transpose_w(const float* __restrict__ W,
                                                     float* __restrict__ Wt) {
  __shared__ float tile[32][33];
  const int bx = blockIdx.x * 32, by = blockIdx.y * 32;
  const int tx = threadIdx.x, ty = threadIdx.y;   // 32 x 8
  #pragma unroll
  for (int i = 0; i < 32; i += 8)
    tile[ty + i][tx] = W[(size_t)(by + ty + i) * IN_F + bx + tx];
  __syncthreads();
  #pragma unroll
  for (int i = 0; i < 32; i += 8)
    Wt[(size_t)(bx + ty + i) * OUT_F + by + tx] = tile[tx][ty + i];
}

// ---------------- Kernel 1: feat_src = feat @ W.T via V_WMMA_F32_16X16X4_F32
// block: 256 threads (8 waves). Tile: BM=32 rows x 256 cols.
// K staged in LDS chunks of 32, double-buffered, async-to-LDS when available.
#define BM 32
#define BK 32
#define ASTRIDE 36    // pad: lane l reads As[l*36 + k] -> distinct banks (period 16)
#define BSTRIDE 264   // pad: rows k and k+2 land on disjoint 16-bank sets

__global__ __launch_bounds__(256) void k_gemm_wmma(const float* __restrict__ feat,
                                                   const float* __restrict__ Wt,
                                                   float* __restrict__ feat_src,
                                                   int N) {
  __shared__ __align__(16) float As[2][BM * ASTRIDE];   // 2 x 4.6 KB
  __shared__ __align__(16) float Bs[2][BK * BSTRIDE];   // 2 x 33.8 KB
  const int t = threadIdx.x;
  const unsigned lane = t & 31u;          // wave32
  const int wave = t >> 5;                // 0..7
  const int rt   = wave & 1;              // row tile (16 rows each)
  const int cg   = wave >> 1;             // col group (64 cols each)
  const int row0 = blockIdx.x * BM;
  const unsigned hi  = lane >> 4;         // 0: lanes 0-15, 1: lanes 16-31
  const unsigned l15 = lane & 15u;
  const bool fullA = (row0 + BM <= N);    // block-uniform

  // ---- staging helpers -------------------------------------------------
  auto stageA = [&](int kc, int buf) {
    const int r  = t >> 3;
    const int c4 = (t & 7) << 2;
    const float* g = feat + (size_t)(row0 + r) * IN_F + kc + c4;
    float* l = &As[buf][r * ASTRIDE + c4];
#ifdef HAVE_ASYNC_LDS
    if (fullA) {
      GASYNC_B128(g, l);
      return;
    }
#endif
    float4 v = make_float4(0.f, 0.f, 0.f, 0.f);
    if (row0 + r < N) v = *(const float4*)g;
    *(float4*)l = v;
  };

  auto stageB = [&](int kc, int buf) {
#ifdef HAVE_ASYNC_LDS
    #pragma unroll
    for (int i = 0; i < 8; ++i) {
      const int idx = t + (i << 8);
      const int kr  = idx >> 6;
      const int j4  = (idx & 63) << 2;
      const float* g = Wt + (size_t)(kc + kr) * OUT_F + j4;
      float* l = &Bs[buf][kr * BSTRIDE + j4];
      GASYNC_B128(g, l);
    }
#else
    float4 tmp[8];
    #pragma unroll
    for (int i = 0; i < 8; ++i) {
      const int idx = t + (i << 8);
      const int kr  = idx >> 6;
      const int j4  = (idx & 63) << 2;
      tmp[i] = *(const float4*)(Wt + (size_t)(kc + kr) * OUT_F + j4);
    }
    #pragma unroll
    for (int i = 0; i < 8; ++i) {
      const int idx = t + (i << 8);
      const int kr  = idx >> 6;
      const int j4  = (idx & 63) << 2;
      *(float4*)(&Bs[buf][kr * BSTRIDE + j4]) = tmp[i];
    }
#endif
  };

  // ---- main loop: double-buffered K chunks -----------------------------
  v8f acc[4] = {};                        // 4 col-tiles of 16, C = 8 VGPRs each
  int buf = 0;
  stageA(0, 0);
  stageB(0, 0);
#ifdef HAVE_ASYNC_LDS
  __builtin_amdgcn_s_wait_asynccnt(0);
#endif
  __syncthreads();

  for (int kc = 0; kc < IN_F; kc += BK) {
    const int nxt = kc + BK;
    if (nxt < IN_F) {                     // prefetch next chunk into other buffer
      stageA(nxt, buf ^ 1);
      stageB(nxt, buf ^ 1);
    }

    #pragma unroll
    for (int kk = 0; kk < BK; kk += 4) {
      // A 16x4 f32 fragment (ISA layout: VGPR0 = K0|K2 across lane halves)
      const int kb = kk + (hi << 1);
      const int arow = rt * 16 + l15;
      v2f a;
      a.x = As[buf][arow * ASTRIDE + kb];
      a.y = As[buf][arow * ASTRIDE + kb + 1];
      #pragma unroll
      for (int ct = 0; ct < 4; ++ct) {
        const int j0 = cg * 64 + ct * 16;
        // B 4x16 fragment (VGPR0 = row K0|K2, VGPR1 = row K1|K3)
        v2f b;
        b.x = Bs[buf][kb * BSTRIDE + j0 + l15];
        b.y = Bs[buf][(kb + 1) * BSTRIDE + j0 + l15];
        acc[ct] = __builtin_amdgcn_wmma_f32_16x16x4_f32(
            false, a, false, b, (short)0, acc[ct], false, false);
      }
    }

#ifdef HAVE_ASYNC_LDS
    __builtin_amdgcn_s_wait_asynccnt(0);  // prefetch into buf^1 complete
#endif
    __syncthreads();
    buf ^= 1;
  }

  // C layout: VGPR i -> rows i (lanes 0-15) and i+8 (lanes 16-31), col = lane&15
  #pragma unroll
  for (int ct = 0; ct < 4; ++ct) {
    const int j0 = cg * 64 + ct * 16;
    #pragma unroll
    for (int i = 0; i < 8; ++i) {
      const int grow = row0 + rt * 16 + i + (hi << 3);
      if (grow < N) feat_src[(size_t)grow * OUT_F + j0 + l15] = acc[ct][i];
    }
  }
}

// ---------------- Kernel 2: el/er = per-head dot(feat_src[n], attn_{l,r}); one wave per node
__global__ __launch_bounds__(256) void k_attn_dots(const float* __restrict__ feat_src,
                                                   const float* __restrict__ attn_l,
                                                   const float* __restrict__ attn_r,
                                                   float* __restrict__ el,
                                                   float* __restrict__ er, int N) {
  const int wave = threadIdx.x >> 5;
  const unsigned lane = threadIdx.x & 31u;
  const int n = blockIdx.x * 8 + wave;
  if (n >= N) return;
  const float* row = feat_src + (size_t)n * OUT_F;
  const int base = lane * 8;              // 8 contiguous per lane; head = lane>>3
  float sl = 0.f, sr = 0.f;
  #pragma unroll
  for (int i = 0; i < 8; ++i) {
    const float f = row[base + i];
    sl += f * attn_l[base + i];
    sr += f * attn_r[base + i];
  }
  #pragma unroll
  for (int off = 1; off < 8; off <<= 1) {  // reduce within 8-lane head group
    sl += __shfl_xor(sl, off, 32);
    sr += __shfl_xor(sr, off, 32);
  }
  if ((lane & 7u) == 0) {
    const int h = lane >> 3;
    el[(size_t)n * NHEAD + h] = sl;
    er[(size_t)n * NHEAD + h] = sr;
  }
}

// ---------------- Kernel 3: leaky-relu + online edge-softmax + multi-head SpMM
// one block (256 thr) per dst node; thread t -> (h = t>>6, d = t&63)
#define CHUNK 32
__global__ __launch_bounds__(256) void k_gat_softmax_spmm(
    const int* __restrict__ row_ptr, const int* __restrict__ col_ind,
    const float* __restrict__ feat_src, const float* __restrict__ el,
    const float* __restrict__ er, float* __restrict__ out, int N) {
  __shared__ float sE[CHUNK * NHEAD];
  __shared__ int   sSrc[CHUNK];
  const int n = blockIdx.x;
  const int t = threadIdx.x;
  const int h = t >> 6;
  const int start = row_ptr[n];
  const int deg = row_ptr[n + 1] - start;

  float m = -3.402823466e+38f;   // running max
  float den = 0.f, accv = 0.f;   // running denom / numerator (flash-style)

  for (int e0 = 0; e0 < deg; e0 += CHUNK) {
    const int cnt = (deg - e0) < CHUNK ? (deg - e0) : CHUNK;
    if (t < cnt) {
      const int s = col_ind[start + e0 + t];
      sSrc[t] = s;
      __builtin_prefetch(feat_src + (size_t)s * OUT_F, 0, 0);  // global_prefetch
    }
    if (t < cnt * NHEAD) {
      const int e = t >> 2, hh = t & 3;
      const int s = col_ind[start + e0 + e];
      const float x = er[(size_t)s * NHEAD + hh] + el[(size_t)n * NHEAD + hh];
      sE[t] = x > 0.f ? x : NEG_SLOPE * x;
    }
    __syncthreads();
    for (int e = 0; e < cnt; ++e) {
      const float s = sE[e * NHEAD + h];
      const float mn = fmaxf(m, s);
      const float scale = expf(m - mn);   // first iter: exp(-inf)=0
      const float p = expf(s - mn);
      den  = den * scale + p;
      accv = accv * scale + p * feat_src[(size_t)sSrc[e] * OUT_F + t];
      m = mn;
    }
    __syncthreads();
  }
  out[(size_t)n * OUT_F + t] = accv / den;
}

// ---------------- launcher
extern "C" void kernel_launch(void* const* d_in, const int* in_sizes, int n_in,
                              void* d_out, int out_size, void* d_ws, size_t ws_size,
                              hipStream_t stream) {
  const int*   row_ptr = (const int*)d_in[0];   // N+1
  const int*   col_ind = (const int*)d_in[1];   // E
  const float* feat    = (const float*)d_in[2]; // N*256
  const float* W       = (const float*)d_in[3]; // 256*256
  const float* attn_l  = (const float*)d_in[4]; // 256
  const float* attn_r  = (const float*)d_in[5]; // 256
  const int N = in_sizes[0] - 1;

  // workspace layout (floats): Wt | el | er | feat_src
  float* ws = (float*)d_ws;
  float* Wt = ws;                                   // 65536
  float* el = Wt + (size_t)OUT_F * IN_F;            // N*4
  float* er = el + (size_t)N * NHEAD;               // N*4
  float* feat_src = er + (size_t)N * NHEAD;         // N*256

  float* out = (float*)d_out;

  k_transpose_w<<<dim3(8, 8), dim3(32, 8), 0, stream>>>(W, Wt);
  k_gemm_wmma<<<(N + BM - 1) / BM, 256, 0, stream>>>(feat, Wt, feat_src, N);
  k_attn_dots<<<(N + 7) / 8, 256, 0, stream>>>(feat_src, attn_l, attn_r, el, er, N);
  k_gat_softmax_spmm<<<N, 256, 0, stream>>>(row_ptr, col_ind, feat_src, el, er, out, N);
}